// GATModel_19662360281715
// MI455X (gfx1250) — compile-verified
//
#include <hip/hip_runtime.h>
#include <hip/hip_bf16.h>
#include <stdint.h>

typedef float v2f __attribute__((ext_vector_type(2)));
typedef float v8f __attribute__((ext_vector_type(8)));

#define FIN  128
#define HC1  256   // HEADS*HID = 4*64
#define OUTC 16

// ---------- helpers: monotone float<->uint mapping for atomicMax-based segment max ----------
__device__ __forceinline__ unsigned flipf(float f) {
  unsigned u = __float_as_uint(f);
  return (u & 0x80000000u) ? ~u : (u ^ 0x80000000u);
}
__device__ __forceinline__ float unflipf(unsigned u) {
  return (u & 0x80000000u) ? __uint_as_float(u ^ 0x80000000u) : __uint_as_float(~u);
}

// ---------- CDNA5 async global->LDS copy (ASYNCcnt path), 16B per lane ----------
__device__ __forceinline__ void async_g2l_b128(const void* gptr, void* lptr) {
  const uint32_t lds_off = (uint32_t)(uintptr_t)lptr;   // low 32 bits of generic addr = LDS offset
  const uint64_t gaddr   = (uint64_t)(uintptr_t)gptr;
  asm volatile("global_load_async_to_lds_b128 %0, %1, off"
               :: "v"(lds_off), "v"(gaddr) : "memory");
}
__device__ __forceinline__ void async_wait0() {
  asm volatile("s_wait_asynccnt 0x0" ::: "memory");
}

// ---------- generic fill ----------
__global__ void gat_fill_u32(unsigned* __restrict__ p, unsigned v, long long n) {
  long long i = (long long)blockIdx.x * 256 + threadIdx.x;
  if (i < n) p[i] = v;
}

// ---------- pair-interleave W [K,Ncols] -> Wsw [K/2][Ncols][2] so B fragments are contiguous b64 ----------
__global__ void gat_swizzleW(const float* __restrict__ W, float* __restrict__ Wsw,
                             int K, int Ncols) {
  long long i = (long long)blockIdx.x * 256 + threadIdx.x;  // over (K/2)*Ncols
  const long long total = (long long)(K / 2) * Ncols;
  if (i >= total) return;
  const int kp = (int)(i / Ncols);
  const int n  = (int)(i - (long long)kp * Ncols);
  float2 v;
  v.x = W[(size_t)(2 * kp) * Ncols + n];
  v.y = W[(size_t)(2 * kp + 1) * Ncols + n];
  *(float2*)(Wsw + i * 2) = v;
}

// ---------- GEMM1: h1[N,256] = x[N,128] @ W1[128,256], fp32 WMMA 16x16x4 ----------
__launch_bounds__(256)
__global__ void gat_gemm1(const float* __restrict__ x, const float* __restrict__ Wsw,
                          float* __restrict__ h1, int N) {
  __shared__ __align__(16) float ldsW[32 * 512];  // 64KB: 32 kpairs x 256 cols x 2
  const int tid  = threadIdx.x;
  const int wave = tid >> 5;
  const int lane = tid & 31;
  const int half = lane >> 4;
  const int l15  = lane & 15;
  const int rowBase = blockIdx.x * 128 + wave * 16;

  v8f acc[16];
#pragma unroll
  for (int t = 0; t < 16; ++t)
#pragma unroll
    for (int r = 0; r < 8; ++r) acc[t][r] = 0.0f;

  int rowA = rowBase + l15;
  if (rowA >= N) rowA = N - 1;  // clamp: keeps EXEC full for WMMA; stores masked later

  for (int phase = 0; phase < 2; ++phase) {
    __syncthreads();
    {
      // async stage 64KB of pre-swizzled W into LDS (identity copy, 16B/lane)
      const float* gsrc = Wsw + (size_t)phase * 16384;
      for (int i = tid; i < 4096; i += 256)
        async_g2l_b128(gsrc + i * 4, ldsW + i * 4);
      async_wait0();
    }
    __syncthreads();

    for (int ks = 0; ks < 16; ++ks) {
      const int kg = phase * 64 + ks * 4 + 2 * half;   // global K for A
      const float2 av = *(const float2*)(x + (size_t)rowA * FIN + kg);
      v2f a; a.x = av.x; a.y = av.y;
      const float* bbase = ldsW + (ks * 2 + half) * 512 + l15 * 2;
#pragma unroll
      for (int nt = 0; nt < 16; ++nt) {
        const float2 bv = *(const float2*)(bbase + nt * 32);  // single ds_load_b64
        v2f b; b.x = bv.x; b.y = bv.y;
        acc[nt] = __builtin_amdgcn_wmma_f32_16x16x4_f32(
            false, a, false, b, (short)0, acc[nt], false, false);
      }
    }
  }

#pragma unroll
  for (int nt = 0; nt < 16; ++nt)
#pragma unroll
    for (int r = 0; r < 8; ++r) {
      const int row = rowBase + r + 8 * half;
      if (row < N) h1[(size_t)row * HC1 + nt * 16 + l15] = acc[nt][r];
    }
}

// ---------- per-node attention dots, layer 1 (H=4, C=64) ----------
__global__ void gat_att1(const float* __restrict__ h1, const float* __restrict__ att_src,
                         const float* __restrict__ att_dst, float* __restrict__ asrc,
                         float* __restrict__ adst, int N) {
  long long i = (long long)blockIdx.x * 256 + threadIdx.x;
  if (i >= (long long)N * 4) return;
  const int n = (int)(i >> 2), h = (int)(i & 3);
  const float* hp = h1 + (size_t)n * HC1 + h * 64;
  const float* as = att_src + h * 64;
  const float* ad = att_dst + h * 64;
  float s = 0.f, d = 0.f;
#pragma unroll 8
  for (int c = 0; c < 64; ++c) { const float v = hp[c]; s += v * as[c]; d += v * ad[c]; }
  asrc[i] = s; adst[i] = d;
}

// ---------- per-node attention dots, layer 2 (H=1, C=16) ----------
__global__ void gat_att2(const float* __restrict__ h2, const float* __restrict__ att_src,
                         const float* __restrict__ att_dst, float* __restrict__ asrc,
                         float* __restrict__ adst, int N) {
  long long n = (long long)blockIdx.x * 256 + threadIdx.x;
  if (n >= N) return;
  const float* hp = h2 + (size_t)n * OUTC;
  float s = 0.f, d = 0.f;
#pragma unroll
  for (int c = 0; c < OUTC; ++c) { const float v = hp[c]; s += v * att_src[c]; d += v * att_dst[c]; }
  asrc[n] = s; adst[n] = d;
}

// ---------- edge pass 1: e = leakyrelu(a_src[src]+a_dst[dst]); segment max via atomicMax ----------
__global__ void gat_epass1(const int* __restrict__ src, const int* __restrict__ dst,
                           const float* __restrict__ asrc, const float* __restrict__ adst,
                           float* __restrict__ ev, unsigned* __restrict__ mu,
                           int E, int Hshift) {
  long long i = (long long)blockIdx.x * 256 + threadIdx.x;
  const long long total = (long long)E << Hshift;
  if (i >= total) return;
  const int e = (int)(i >> Hshift);
  const int h = (int)(i & ((1 << Hshift) - 1));
  const int H = 1 << Hshift;
  const int s = src[e], d = dst[e];
  float v = asrc[(size_t)s * H + h] + adst[(size_t)d * H + h];
  v = v > 0.f ? v : 0.2f * v;
  ev[i] = v;
  atomicMax(&mu[(size_t)d * H + h], flipf(v));
}

// ---------- edge pass 2: ex = exp(e - m[dst]); segment sum ----------
__global__ void gat_epass2(const int* __restrict__ dst, float* __restrict__ ev,
                           const unsigned* __restrict__ mu, float* __restrict__ ssum,
                           int E, int Hshift) {
  long long i = (long long)blockIdx.x * 256 + threadIdx.x;
  const long long total = (long long)E << Hshift;
  if (i >= total) return;
  const int e = (int)(i >> Hshift);
  const int h = (int)(i & ((1 << Hshift) - 1));
  const int H = 1 << Hshift;
  const int d = dst[e];
  const float m  = unflipf(mu[(size_t)d * H + h]);
  const float ex = expf(ev[i] - m);
  ev[i] = ex;
  atomicAdd(&ssum[(size_t)d * H + h], ex);
}

// ---------- edge pass 3: out[dst] += h[src] * alpha  (float4 gather, 4 scalar f32 atomics) ----------
__global__ void gat_epass3(const int* __restrict__ src, const int* __restrict__ dst,
                           const float* __restrict__ ev, const float* __restrict__ ssum,
                           const float* __restrict__ feat, float* __restrict__ outp,
                           int E, int perEshift, int rHshift, int H) {
  long long i = (long long)blockIdx.x * 256 + threadIdx.x;
  const long long total = (long long)E << perEshift;
  if (i >= total) return;
  const int e = (int)(i >> perEshift);
  const int r = (int)(i & ((1 << perEshift) - 1));
  const int h = r >> rHshift;
  const int s = src[e], d = dst[e];
  const float alpha = ev[(size_t)e * H + h] / (ssum[(size_t)d * H + h] + 1e-16f);
  const int FS = 4 << perEshift;  // features per node = H*C
  const float4 f = *(const float4*)(feat + (size_t)s * FS + r * 4);
  float* o = outp + (size_t)d * FS + r * 4;
  atomicAdd(o + 0, f.x * alpha);
  atomicAdd(o + 1, f.y * alpha);
  atomicAdd(o + 2, f.z * alpha);
  atomicAdd(o + 3, f.w * alpha);
}

// ---------- GEMM2: h2[N,16] = relu(out1 + b1) @ W2[256,16], bias+relu fused into A load ----------
__launch_bounds__(256)
__global__ void gat_gemm2(const float* __restrict__ A, const float* __restrict__ W2sw,
                          const float* __restrict__ b1, float* __restrict__ h2, int N) {
  __shared__ __align__(16) float ldsW[128 * 32];  // 16KB: 128 kpairs x 16 cols x 2
  __shared__ __align__(16) float ldsB[HC1];
  const int tid = threadIdx.x;
  for (int i = tid; i < 1024; i += 256)
    async_g2l_b128(W2sw + i * 4, ldsW + i * 4);
  if (tid < 64)
    async_g2l_b128(b1 + tid * 4, ldsB + tid * 4);
  async_wait0();
  __syncthreads();

  const int wave = tid >> 5;
  const int lane = tid & 31;
  const int half = lane >> 4;
  const int l15  = lane & 15;
  const int rowBase = blockIdx.x * 128 + wave * 16;
  int rowA = rowBase + l15;
  if (rowA >= N) rowA = N - 1;

  v8f acc;
#pragma unroll
  for (int r = 0; r < 8; ++r) acc[r] = 0.0f;

  for (int ks = 0; ks < 64; ++ks) {
    const int k = ks * 4 + 2 * half;
    const float2 av = *(const float2*)(A + (size_t)rowA * HC1 + k);
    v2f a;
    a.x = fmaxf(av.x + ldsB[k],     0.f);
    a.y = fmaxf(av.y + ldsB[k + 1], 0.f);
    const float2 bv = *(const float2*)(ldsW + (ks * 2 + half) * 32 + l15 * 2);
    v2f b; b.x = bv.x; b.y = bv.y;
    acc = __builtin_amdgcn_wmma_f32_16x16x4_f32(false, a, false, b, (short)0, acc, false, false);
  }

#pragma unroll
  for (int r = 0; r < 8; ++r) {
    const int row = rowBase + r + 8 * half;
    if (row < N) h2[(size_t)row * OUTC + l15] = acc[r];
  }
}

// ---------- final bias ----------
__global__ void gat_bias2(float* __restrict__ out, const float* __restrict__ b2, int N) {
  long long i = (long long)blockIdx.x * 256 + threadIdx.x;
  if (i >= (long long)N * OUTC) return;
  out[i] += b2[i & (OUTC - 1)];
}

extern "C" void kernel_launch(void* const* d_in, const int* in_sizes, int n_in,
                              void* d_out, int out_size, void* d_ws, size_t ws_size,
                              hipStream_t stream) {
  const float* x   = (const float*)d_in[0];
  const int*   ei  = (const int*)d_in[1];
  const float* W1  = (const float*)d_in[2];
  const float* as1 = (const float*)d_in[3];
  const float* ad1 = (const float*)d_in[4];
  const float* b1  = (const float*)d_in[5];
  const float* W2  = (const float*)d_in[6];
  const float* as2 = (const float*)d_in[7];
  const float* ad2 = (const float*)d_in[8];
  const float* b2  = (const float*)d_in[9];
  float* out = (float*)d_out;

  const int N = in_sizes[0] / FIN;
  const int E = in_sizes[1] / 2;
  const int* src = ei;
  const int* dst = ei + E;

  // workspace layout (~122 MB + 144KB)
  float*    h1   = (float*)d_ws;                       // N*256
  float*    out1 = h1 + (size_t)N * HC1;               // N*256
  float*    aS1  = out1 + (size_t)N * HC1;             // N*4
  float*    aD1  = aS1 + (size_t)N * 4;                // N*4
  unsigned* mu1  = (unsigned*)(aD1 + (size_t)N * 4);   // N*4
  float*    s1   = (float*)(mu1 + (size_t)N * 4);      // N*4
  float*    ev   = s1 + (size_t)N * 4;                 // E*4 (reused by layer 2)
  float*    h2   = ev + (size_t)E * 4;                 // N*16
  float*    aS2  = h2 + (size_t)N * OUTC;              // N
  float*    aD2  = aS2 + (size_t)N;                    // N
  unsigned* mu2  = (unsigned*)(aD2 + (size_t)N);       // N
  float*    s2   = (float*)(mu2 + (size_t)N);          // N
  float*    W1sw = s2 + (size_t)N;                     // 128*256 pair-interleaved
  float*    W2sw = W1sw + FIN * HC1;                   // 256*16 pair-interleaved

  const unsigned NEGINF_FLIP = 0x007FFFFFu;  // flipf(-inf)
  auto nb = [](long long n) { return (unsigned)((n + 255) / 256); };

  // weight pre-swizzle (tiny, once per call)
  gat_swizzleW<<<nb((long long)(FIN / 2) * HC1), 256, 0, stream>>>(W1, W1sw, FIN, HC1);
  gat_swizzleW<<<nb((long long)(HC1 / 2) * OUTC), 256, 0, stream>>>(W2, W2sw, HC1, OUTC);

  // init accumulators / segment-max seeds
  gat_fill_u32<<<nb((long long)N * HC1), 256, 0, stream>>>((unsigned*)out1, 0u, (long long)N * HC1);
  gat_fill_u32<<<nb((long long)N * 4), 256, 0, stream>>>(mu1, NEGINF_FLIP, (long long)N * 4);
  gat_fill_u32<<<nb((long long)N * 4), 256, 0, stream>>>((unsigned*)s1, 0u, (long long)N * 4);
  gat_fill_u32<<<nb(N), 256, 0, stream>>>(mu2, NEGINF_FLIP, N);
  gat_fill_u32<<<nb(N), 256, 0, stream>>>((unsigned*)s2, 0u, N);
  gat_fill_u32<<<nb((long long)N * OUTC), 256, 0, stream>>>((unsigned*)out, 0u, (long long)N * OUTC);

  // layer 1
  gat_gemm1<<<(N + 127) / 128, 256, 0, stream>>>(x, W1sw, h1, N);
  gat_att1<<<nb((long long)N * 4), 256, 0, stream>>>(h1, as1, ad1, aS1, aD1, N);
  gat_epass1<<<nb((long long)E * 4), 256, 0, stream>>>(src, dst, aS1, aD1, ev, mu1, E, 2);
  gat_epass2<<<nb((long long)E * 4), 256, 0, stream>>>(dst, ev, mu1, s1, E, 2);
  gat_epass3<<<nb((long long)E * 64), 256, 0, stream>>>(src, dst, ev, s1, h1, out1, E, 6, 4, 4);

  // layer 2 (bias1 + relu fused into GEMM2 A-operand load)
  gat_gemm2<<<(N + 127) / 128, 256, 0, stream>>>(out1, W2sw, b1, h2, N);
  gat_att2<<<nb(N), 256, 0, stream>>>(h2, as2, ad2, aS2, aD2, N);
  gat_epass1<<<nb(E), 256, 0, stream>>>(src, dst, aS2, aD2, ev, mu2, E, 0);
  gat_epass2<<<nb(E), 256, 0, stream>>>(dst, ev, mu2, s2, E, 0);
  gat_epass3<<<nb((long long)E * 4), 256, 0, stream>>>(src, dst, ev, s2, h2, out, E, 2, 2, 1);
  gat_bias2<<<nb((long long)N * OUTC), 256, 0, stream>>>(out, b2, N);
}